// soft_dice_cldice_54984171323930
// MI455X (gfx1250) — compile-verified
//
#include <hip/hip_runtime.h>
#include <hip/hip_bf16.h>

#define NX 96
#define NY 96
#define NZ 96
#define VOL (NX * NY * NZ)          // 884736, divisible by 64
#define ITERS 25
#define ALPHA 0.1f
#define SMOOTH 1.0f
#define RED_BLOCKS 240

typedef __attribute__((ext_vector_type(2))) float v2f;
typedef __attribute__((ext_vector_type(8))) float v8f;

// ---------------------------------------------------------------------------
// 7-point cross min ( = min(minpool_z, minpool_y, minpool_x), SAME padding )
// ---------------------------------------------------------------------------
__global__ __launch_bounds__(256) void k_erode(const float* __restrict__ in,
                                               float* __restrict__ out) {
    int x = blockIdx.x * 32 + threadIdx.x;
    int y = blockIdx.y * 4 + threadIdx.y;
    int z = blockIdx.z * 2 + threadIdx.z;
    int idx = (z * NY + y) * NX + x;
    float m = in[idx];
    if (x > 0)      m = fminf(m, in[idx - 1]);
    if (x < NX - 1) m = fminf(m, in[idx + 1]);
    if (y > 0)      m = fminf(m, in[idx - NX]);
    if (y < NY - 1) m = fminf(m, in[idx + NX]);
    if (z > 0)      m = fminf(m, in[idx - NX * NY]);
    if (z < NZ - 1) m = fminf(m, in[idx + NX * NY]);
    out[idx] = m;
}

// ---------------------------------------------------------------------------
// D = 27-point max of `er`; delta = relu(ref - D);
// init: skel = delta; else: skel += relu(delta - skel*delta)
// ---------------------------------------------------------------------------
__global__ __launch_bounds__(256) void k_open_update(const float* __restrict__ er,
                                                     const float* __restrict__ ref,
                                                     float* __restrict__ skel,
                                                     int init) {
    int x = blockIdx.x * 32 + threadIdx.x;
    int y = blockIdx.y * 4 + threadIdx.y;
    int z = blockIdx.z * 2 + threadIdx.z;
    int idx = (z * NY + y) * NX + x;

    float d = -3.4e38f;
    #pragma unroll
    for (int dz = -1; dz <= 1; ++dz) {
        int zz = z + dz;
        if ((unsigned)zz > (unsigned)(NZ - 1)) continue;
        #pragma unroll
        for (int dy = -1; dy <= 1; ++dy) {
            int yy = y + dy;
            if ((unsigned)yy > (unsigned)(NY - 1)) continue;
            const float* row = er + (zz * NY + yy) * NX;
            #pragma unroll
            for (int dx = -1; dx <= 1; ++dx) {
                int xx = x + dx;
                if ((unsigned)xx > (unsigned)(NX - 1)) continue;
                d = fmaxf(d, row[xx]);
            }
        }
    }
    float delta = fmaxf(ref[idx] - d, 0.0f);
    if (init) {
        skel[idx] = delta;
    } else {
        float s = skel[idx];
        skel[idx] = s + fmaxf(delta - s * delta, 0.0f);
    }
}

// ---------------------------------------------------------------------------
// WMMA f32 reduction: per-block partial sums of (s*w) and (s).
// Each wave feeds 64 f32 values per v_wmma_f32_16x16x4_f32 (B = ones);
// accumulator grand-total = sum(C)/16 exactly. Wave-uniform loop => EXEC all 1s.
// ---------------------------------------------------------------------------
__global__ __launch_bounds__(256) void k_reduce(const float* __restrict__ s,
                                                const float* __restrict__ w,
                                                float* __restrict__ outSW,
                                                float* __restrict__ outS) {
    const int lane  = threadIdx.x & 31;
    const int gwave = blockIdx.x * 8 + (threadIdx.x >> 5);
    const int nwave = gridDim.x * 8;
    const int nchunk = VOL / 64;

    v8f cSW = {}; v8f cS = {};
    v2f ones; ones.x = 1.0f; ones.y = 1.0f;

    for (int chunk = gwave; chunk < nchunk; chunk += nwave) {
        int base = chunk * 64;
        float s0 = s[base + lane];
        float s1 = s[base + 32 + lane];
        float w0 = w[base + lane];
        float w1 = w[base + 32 + lane];
        v2f aS;  aS.x = s0;       aS.y = s1;
        v2f aSW; aSW.x = s0 * w0; aSW.y = s1 * w1;
        cSW = __builtin_amdgcn_wmma_f32_16x16x4_f32(false, aSW, false, ones,
                                                    (short)0, cSW, false, false);
        cS  = __builtin_amdgcn_wmma_f32_16x16x4_f32(false, aS,  false, ones,
                                                    (short)0, cS,  false, false);
    }

    float tSW = 0.0f, tS = 0.0f;
    #pragma unroll
    for (int v = 0; v < 8; ++v) { tSW += cSW[v]; tS += cS[v]; }

    __shared__ float l0[256];
    __shared__ float l1[256];
    l0[threadIdx.x] = tSW;
    l1[threadIdx.x] = tS;
    __syncthreads();
    for (int off = 128; off > 0; off >>= 1) {
        if ((int)threadIdx.x < off) {
            l0[threadIdx.x] += l0[threadIdx.x + off];
            l1[threadIdx.x] += l1[threadIdx.x + off];
        }
        __syncthreads();
    }
    if (threadIdx.x == 0) {
        outSW[blockIdx.x] = l0[0] * (1.0f / 16.0f);
        outS[blockIdx.x]  = l1[0] * (1.0f / 16.0f);
    }
}

// ---------------------------------------------------------------------------
// Final combine: partials layout [S0|S1|S2|S3] each RED_BLOCKS long.
// ---------------------------------------------------------------------------
__global__ __launch_bounds__(256) void k_final(const float* __restrict__ p, int nblk,
                                               const float* __restrict__ dice,
                                               float* __restrict__ out) {
    float a0 = 0.f, a1 = 0.f, a2 = 0.f, a3 = 0.f;
    for (int i = threadIdx.x; i < nblk; i += blockDim.x) {
        a0 += p[i];
        a1 += p[nblk + i];
        a2 += p[2 * nblk + i];
        a3 += p[3 * nblk + i];
    }
    __shared__ float l[4][256];
    l[0][threadIdx.x] = a0; l[1][threadIdx.x] = a1;
    l[2][threadIdx.x] = a2; l[3][threadIdx.x] = a3;
    __syncthreads();
    for (int off = 128; off > 0; off >>= 1) {
        if ((int)threadIdx.x < off) {
            l[0][threadIdx.x] += l[0][threadIdx.x + off];
            l[1][threadIdx.x] += l[1][threadIdx.x + off];
            l[2][threadIdx.x] += l[2][threadIdx.x + off];
            l[3][threadIdx.x] += l[3][threadIdx.x + off];
        }
        __syncthreads();
    }
    if (threadIdx.x == 0) {
        float S0 = l[0][0], S1 = l[1][0], S2 = l[2][0], S3 = l[3][0];
        float tprec = (S0 + SMOOTH) / (S1 + SMOOTH);
        float tsens = (S2 + SMOOTH) / (S3 + SMOOTH);
        float cl = 1.0f - 2.0f * (tprec * tsens) / (tprec + tsens);
        out[0] = (1.0f - ALPHA) * dice[0] + ALPHA * cl;
    }
}

// ---------------------------------------------------------------------------
static void run_image(const float* img, const float* other,
                      float* A, float* B, float* skel,
                      float* pSW, float* pS, hipStream_t stream) {
    dim3 blk(32, 4, 2);
    dim3 grd(NX / 32, NY / 4, NZ / 2);
    // step 0: A1 = erode(img); skel = relu(img - dilate(A1))
    k_erode<<<grd, blk, 0, stream>>>(img, B);
    k_open_update<<<grd, blk, 0, stream>>>(B, img, skel, 1);
    float* cur = B;
    float* nxt = A;
    for (int it = 1; it <= ITERS; ++it) {
        // A_{it+1} = erode(A_it); delta from dilate(A_{it+1}) vs A_it
        k_erode<<<grd, blk, 0, stream>>>(cur, nxt);
        k_open_update<<<grd, blk, 0, stream>>>(nxt, cur, skel, 0);
        float* t = cur; cur = nxt; nxt = t;
    }
    k_reduce<<<RED_BLOCKS, 256, 0, stream>>>(skel, other, pSW, pS);
}

extern "C" void kernel_launch(void* const* d_in, const int* in_sizes, int n_in,
                              void* d_out, int out_size, void* d_ws, size_t ws_size,
                              hipStream_t stream) {
    const float* y_true = (const float*)d_in[0];
    const float* y_pred = (const float*)d_in[1];
    const float* dice   = (const float*)d_in[2];

    float* A     = (float*)d_ws;
    float* B     = A + VOL;
    float* skel  = B + VOL;
    float* parts = skel + VOL;        // 4 * RED_BLOCKS floats

    // skel(y_pred) vs y_true -> S0, S1
    run_image(y_pred, y_true, A, B, skel, parts, parts + RED_BLOCKS, stream);
    // skel(y_true) vs y_pred -> S2, S3
    run_image(y_true, y_pred, A, B, skel, parts + 2 * RED_BLOCKS,
              parts + 3 * RED_BLOCKS, stream);

    k_final<<<1, 256, 0, stream>>>(parts, RED_BLOCKS, dice, (float*)d_out);
}